// UpsampleUpFIRDn_1580547967880
// MI455X (gfx1250) — compile-verified
//
#include <hip/hip_runtime.h>

// upfirdn2d up=2, 4x4 FIR, pads (2,1):  in  [BC,256,256] f32  ->  out [BC,511,511] f32
// Memory-bound (~670 MB traffic, ~29 us floor at 23.3 TB/s). CDNA5 path used:
// async global->LDS staging (global_load_async_to_lds_b32) with double buffering
// and s_wait_asynccnt pipelining; wave32 coalesced stores.

#define H_IN   256
#define W_IN   256
#define H_OUT  511
#define W_OUT  511
#define QR_TILE 16              // quad rows per tile   (=> 32 output rows would be 2*16; rows 2r,2r+1)
#define QC_TILE 64              // quad cols per tile   (=> 128 output cols)
#define LR (QR_TILE + 2)        // 18 LDS rows  (halo 1 top/bottom)
#define LC (QC_TILE + 2)        // 66 LDS cols  (halo 1 left/right)
#define LDS_N (LR * LC)         // 1188 floats per buffer
#define NTHREADS 256
#define COL_TILES (W_IN / QC_TILE)  // 4

typedef __attribute__((address_space(3))) float lds_f;

// Issue async global->LDS loads for one 18x66 input tile (zero-fill OOB halo).
__device__ __forceinline__ void issue_tile_loads(const float* __restrict__ xplane,
                                                 float* buf, int R0, int C0, int tid)
{
    lds_f* b3 = (lds_f*)buf;
#pragma unroll
    for (int it = 0; it < (LDS_N + NTHREADS - 1) / NTHREADS; ++it) {
        int idx = tid + it * NTHREADS;
        if (idx < LDS_N) {
            int lr = idx / LC;
            int lc = idx - lr * LC;
            int gr = R0 - 1 + lr;
            int gc = C0 - 1 + lc;
            if ((unsigned)gr < (unsigned)H_IN && (unsigned)gc < (unsigned)W_IN) {
                unsigned voff = (unsigned)(gr * W_IN + gc) * 4u;
                // dst LDS byte-offset (AS3 ptr in VGPR), 32-bit vaddr offset, 64-bit sgpr base
                asm volatile("global_load_async_to_lds_b32 %0, %1, %2"
                             :
                             : "v"(b3 + idx), "v"(voff), "s"(xplane)
                             : "memory");
            } else {
                buf[idx] = 0.0f;   // zero halo (pad region) with a plain DS store
            }
        }
    }
}

// Compute 4 output quads per thread from the staged LDS tile.
__device__ __forceinline__ void compute_tile(const float* buf,
                                             float* __restrict__ oplane,
                                             const float w[4][4],
                                             int R0, int C0, int tid)
{
    const int qc  = tid & (QC_TILE - 1);   // 0..63
    const int qr0 = tid >> 6;              // 0..3
#pragma unroll
    for (int i = 0; i < QR_TILE / 4; ++i) {
        const int qr = qr0 + i * 4;
        const int r  = R0 + qr;
        const int c  = C0 + qc;
        const float* s = buf + qr * LC + qc;   // s[dr][dc] == x[r-1+dr][c-1+dc]
        const float s00 = s[0],        s01 = s[1],        s02 = s[2];
        const float s10 = s[LC],       s11 = s[LC + 1],   s12 = s[LC + 2];
        const float s20 = s[2 * LC],   s21 = s[2 * LC+1], s22 = s[2 * LC + 2];

        // parity-decomposed 2x2 taps; w[][] is the flipped kernel
        const float v00 = w[0][0]*s00 + w[0][2]*s01 + w[2][0]*s10 + w[2][2]*s11;
        const float v01 = w[0][1]*s01 + w[0][3]*s02 + w[2][1]*s11 + w[2][3]*s12;
        const float v10 = w[1][0]*s10 + w[1][2]*s11 + w[3][0]*s20 + w[3][2]*s21;
        const float v11 = w[1][1]*s11 + w[1][3]*s12 + w[3][1]*s21 + w[3][3]*s22;

        const int oy = 2 * r, ox = 2 * c;
        float* orow = oplane + (size_t)oy * W_OUT + ox;
        orow[0] = v00;
        const bool cok = (ox + 1) < W_OUT;
        if (cok) orow[1] = v01;
        if ((oy + 1) < H_OUT) {
            float* orow2 = orow + W_OUT;
            orow2[0] = v10;
            if (cok) orow2[1] = v11;
        }
    }
}

__global__ __launch_bounds__(NTHREADS)
void upfirdn2x_async_kernel(const float* __restrict__ x,
                            const float* __restrict__ kern,
                            float* __restrict__ out)
{
    __shared__ float tile[2][LDS_N];       // double-buffered staging
    const int tid   = (int)threadIdx.x;
    const int tr    = (int)blockIdx.x;     // 0..15 row-tile of quad grid
    const int plane = (int)blockIdx.y;     // 0..BC-1

    const float* xplane = x   + (size_t)plane * (size_t)(H_IN * W_IN);
    float*       oplane = out + (size_t)plane * ((size_t)H_OUT * (size_t)W_OUT);

    // Flipped 4x4 FIR weights (reference flips before correlation).
    float w[4][4];
#pragma unroll
    for (int i = 0; i < 4; ++i)
#pragma unroll
        for (int j = 0; j < 4; ++j)
            w[i][j] = kern[(3 - i) * 4 + (3 - j)];

    const int R0 = tr * QR_TILE;

    // Pipeline: prefetch tile 0, then {wait, prefetch t+1, compute t}.
    issue_tile_loads(xplane, tile[0], R0, 0, tid);
#pragma unroll
    for (int t = 0; t < COL_TILES; ++t) {
        asm volatile("s_wait_asynccnt 0" ::: "memory");  // this wave's async loads done
        __syncthreads();                                 // all waves' tiles visible
        if (t + 1 < COL_TILES)
            issue_tile_loads(xplane, tile[(t + 1) & 1], R0, (t + 1) * QC_TILE, tid);
        compute_tile(tile[t & 1], oplane, w, R0, t * QC_TILE, tid);
    }
}

extern "C" void kernel_launch(void* const* d_in, const int* in_sizes, int n_in,
                              void* d_out, int out_size, void* d_ws, size_t ws_size,
                              hipStream_t stream)
{
    const float* x    = (const float*)d_in[0];   // [BC,256,256] f32
    const float* kern = (const float*)d_in[1];   // [4,4] f32
    float*       out  = (float*)d_out;           // [BC,511,511] f32

    const int BC = in_sizes[0] / (H_IN * W_IN);  // 8*64 = 512
    dim3 grid((unsigned)(H_IN / QR_TILE) /*16*/, (unsigned)BC, 1);
    dim3 block(NTHREADS, 1, 1);
    upfirdn2x_async_kernel<<<grid, block, 0, stream>>>(x, kern, out);
}